// DiffPoolBatchedGraphLayer_12730283065989
// MI455X (gfx1250) — compile-verified
//
#include <hip/hip_runtime.h>
#include <hip/hip_bf16.h>
#include <math.h>

typedef float v2f __attribute__((ext_vector_type(2)));
typedef float v8f __attribute__((ext_vector_type(8)));

#define DM 128          // feature dim
#define KM 128          // assignment dim
#define NPG 1024        // nodes per graph
#define BG 32           // graphs
#define NKTOT (BG*KM)   // 4096

__device__ __forceinline__ float wave_sum(float v) {
#pragma unroll
    for (int o = 16; o > 0; o >>= 1) v += __shfl_xor(v, o, 32);
    return v;
}
__device__ __forceinline__ float wave_max(float v) {
#pragma unroll
    for (int o = 16; o > 0; o >>= 1) v = fmaxf(v, __shfl_xor(v, o, 32));
    return v;
}

// ---------------- node attention scores: eh, et ----------------
__global__ void node_scores_kernel(const float* __restrict__ h,
                                   const float* __restrict__ Wh,
                                   const float* __restrict__ Wt,
                                   float* __restrict__ eh, float* __restrict__ et, int N) {
    int wave = (blockIdx.x * blockDim.x + threadIdx.x) >> 5;
    int lane = threadIdx.x & 31;
    if (wave >= N) return;
    const float4 hv = *(const float4*)(h + (size_t)wave * DM + lane * 4);
    const float4 wa = *(const float4*)(Wh + lane * 4);
    const float4 wb = *(const float4*)(Wt + lane * 4);
    float sh = hv.x * wa.x + hv.y * wa.y + hv.z * wa.z + hv.w * wa.w;
    float st = hv.x * wb.x + hv.y * wb.y + hv.z * wb.z + hv.w * wb.w;
    sh = wave_sum(sh);
    st = wave_sum(st);
    if (lane == 0) { eh[wave] = tanhf(sh); et[wave] = tanhf(st); }
}

// ---------------- edge logits: e = relu(eh[src]+et[dst]+tanh(r.Wr)) ----------------
__global__ void edge_scores_kernel(const float* __restrict__ r,
                                   const float* __restrict__ Wr,
                                   const float* __restrict__ eh, const float* __restrict__ et,
                                   const int* __restrict__ src, const int* __restrict__ dst,
                                   float* __restrict__ eatt, int E) {
    int e = (blockIdx.x * blockDim.x + threadIdx.x) >> 5;
    int lane = threadIdx.x & 31;
    if (e >= E) return;
    const float4 rv = *(const float4*)(r + (size_t)e * DM + lane * 4);
    const float4 w = *(const float4*)(Wr + lane * 4);
    float s = rv.x * w.x + rv.y * w.y + rv.z * w.z + rv.w * w.w;
    s = wave_sum(s);
    if (lane == 0) {
        float v = tanhf(s) + eh[src[e]] + et[dst[e]];
        eatt[e] = fmaxf(v, 0.0f);
    }
}

// ---------------- segment max over dst (uint trick valid: logits >= 0) ----------------
__global__ void seg_max_kernel(const float* __restrict__ eatt, const int* __restrict__ dst,
                               unsigned int* __restrict__ maxb, int E) {
    int i = blockIdx.x * blockDim.x + threadIdx.x;
    if (i >= E) return;
    atomicMax(&maxb[dst[i]], __float_as_uint(eatt[i]));
}

// ---------------- ex = exp(e - max[dst]); denom[dst] += ex (in-place on eatt) ----------------
__global__ void exp_denom_kernel(float* __restrict__ eatt, const int* __restrict__ dst,
                                 const unsigned int* __restrict__ maxb,
                                 float* __restrict__ den, int E) {
    int i = blockIdx.x * blockDim.x + threadIdx.x;
    if (i >= E) return;
    int d = dst[i];
    float ex = __expf(eatt[i] - __uint_as_float(maxb[d]));
    eatt[i] = ex;
    unsafeAtomicAdd(&den[d], ex);
}

// ---------------- ft[dst] += (ex/denom[dst]) * h[src] ----------------
__global__ void ft_scatter_kernel(const float* __restrict__ h,
                                  const float* __restrict__ eatt, const float* __restrict__ den,
                                  const int* __restrict__ src, const int* __restrict__ dst,
                                  float* __restrict__ ft, int E) {
    int e = (blockIdx.x * blockDim.x + threadIdx.x) >> 5;
    int lane = threadIdx.x & 31;
    if (e >= E) return;
    int s = src[e], d = dst[e];
    float a = eatt[e] / den[d];
    const float4 hv = *(const float4*)(h + (size_t)s * DM + lane * 4);
    float* base = ft + (size_t)d * DM + lane * 4;
    unsafeAtomicAdd(base + 0, a * hv.x);
    unsafeAtomicAdd(base + 1, a * hv.y);
    unsafeAtomicAdd(base + 2, a * hv.z);
    unsafeAtomicAdd(base + 3, a * hv.w);
}

// ---------------- z = ft @ Wfc^T + b; row-normalize; softmax -> S  (WMMA f32 16x16x4) ----------------
__global__ void z_softmax_kernel(const float* __restrict__ ft,
                                 const float* __restrict__ Wfc, const float* __restrict__ bfc,
                                 float* __restrict__ S, int N) {
    __shared__ float zt[16][DM + 4];
    const int m0 = blockIdx.x * 16;           // node row block
    const int w = threadIdx.x >> 5;           // wave id 0..7 -> k column tile
    const int lane = threadIdx.x & 31;
    const int k0 = w * 16;
    const int m = lane & 15;                  // row (A) / col (B,D) within tile
    const int kc = (lane >> 4) << 1;          // K sub-pair: lanes 0-15 -> {0,1}, 16-31 -> {2,3}

    v8f acc = {0.f, 0.f, 0.f, 0.f, 0.f, 0.f, 0.f, 0.f};
#pragma unroll 4
    for (int d0 = 0; d0 < DM; d0 += 4) {
        v2f a = *(const v2f*)(ft + (size_t)(m0 + m) * DM + d0 + kc);          // A: 16x4 of ft
        v2f b = *(const v2f*)(Wfc + (size_t)(k0 + m) * DM + d0 + kc);         // B: 4x16 of Wfc^T
        acc = __builtin_amdgcn_wmma_f32_16x16x4_f32(false, a, false, b, (short)0, acc, false, false);
    }
    const float bias = bfc[k0 + m];
    const int rbase = (lane >> 4) * 8;        // lanes 16-31 hold M = j+8
#pragma unroll
    for (int j = 0; j < 8; j++) zt[j + rbase][k0 + m] = acc[j] + bias;
    __syncthreads();

    // each wave post-processes 2 rows: L2 normalize then softmax over K=128
    for (int rr = 0; rr < 2; rr++) {
        int row = w * 2 + rr;
        float4 z = *(const float4*)(&zt[row][lane * 4]);
        float ss = z.x * z.x + z.y * z.y + z.z * z.z + z.w * z.w;
        ss = wave_sum(ss);
        float inv = 1.0f / fmaxf(sqrtf(ss), 1e-12f);
        z.x *= inv; z.y *= inv; z.z *= inv; z.w *= inv;
        float mx = fmaxf(fmaxf(z.x, z.y), fmaxf(z.z, z.w));
        mx = wave_max(mx);
        float e0 = __expf(z.x - mx), e1 = __expf(z.y - mx);
        float e2 = __expf(z.z - mx), e3 = __expf(z.w - mx);
        float sum = wave_sum(e0 + e1 + e2 + e3);
        float rs = 1.0f / sum;
        float4 o = {e0 * rs, e1 * rs, e2 * rs, e3 * rs};
        *(float4*)(S + (size_t)(m0 + row) * KM + lane * 4) = o;
    }
}

// ---------------- P[dst] += S[src] ----------------
__global__ void p_scatter_kernel(const float* __restrict__ S,
                                 const int* __restrict__ src, const int* __restrict__ dst,
                                 float* __restrict__ P, int E) {
    int e = (blockIdx.x * blockDim.x + threadIdx.x) >> 5;
    int lane = threadIdx.x & 31;
    if (e >= E) return;
    int s = src[e], d = dst[e];
    const float4 sv = *(const float4*)(S + (size_t)s * KM + lane * 4);
    float* base = P + (size_t)d * KM + lane * 4;
    unsafeAtomicAdd(base + 0, sv.x);
    unsafeAtomicAdd(base + 1, sv.y);
    unsafeAtomicAdd(base + 2, sv.z);
    unsafeAtomicAdd(base + 3, sv.w);
}

// ---------------- per-graph pooling GEMMs: h_pool = S^T h ; adj_b = P^T S  (WMMA) ----------------
// grid.x = BG * 8(row tiles) * 2(which); 8 waves = 8 column tiles
__global__ void pool_gemm_kernel(const float* __restrict__ S, const float* __restrict__ P,
                                 const float* __restrict__ h, float* __restrict__ out) {
    const int which = blockIdx.x & 1;               // 0: h_pool, 1: adj
    const int kt = (blockIdx.x >> 1) & 7;           // output row tile (k dim)
    const int b = blockIdx.x >> 4;                  // graph
    const int w = threadIdx.x >> 5;                 // column tile
    const int lane = threadIdx.x & 31;
    const int nb = b * NPG;
    const int k0 = kt * 16, c0 = w * 16;
    const int m = lane & 15;
    const int kc = (lane >> 4) << 1;

    const float* A = which ? P : S;                 // [1024,128] per graph, reduced over n
    const float* Bm = which ? S : h;

    v8f acc = {0.f, 0.f, 0.f, 0.f, 0.f, 0.f, 0.f, 0.f};
    for (int n0 = 0; n0 < NPG; n0 += 4) {
        int r0 = nb + n0 + kc, r1 = r0 + 1;
        v2f a, bb;
        a.x = A[(size_t)r0 * KM + k0 + m];          // A^T tile: A[m][k] = src[n, k0+m]
        a.y = A[(size_t)r1 * KM + k0 + m];
        bb.x = Bm[(size_t)r0 * DM + c0 + m];        // B tile: B[k][n] = src[n_row, c0+n]
        bb.y = Bm[(size_t)r1 * DM + c0 + m];
        acc = __builtin_amdgcn_wmma_f32_16x16x4_f32(false, a, false, bb, (short)0, acc, false, false);
    }
    const int rbase = (lane >> 4) * 8;
    if (which == 0) {
        float* hp = out + (size_t)NKTOT * NKTOT;    // h_pool region
#pragma unroll
        for (int j = 0; j < 8; j++) {
            int k = k0 + j + rbase;
            hp[((size_t)b * KM + k) * DM + c0 + m] = acc[j];
        }
    } else {
#pragma unroll
        for (int j = 0; j < 8; j++) {
            int k = k0 + j + rbase;
            out[(size_t)(b * KM + k) * NKTOT + b * KM + c0 + m] = acc[j];
        }
    }
}

extern "C" void kernel_launch(void* const* d_in, const int* in_sizes, int n_in,
                              void* d_out, int out_size, void* d_ws, size_t ws_size,
                              hipStream_t stream) {
    const float* h    = (const float*)d_in[0];
    const float* r    = (const float*)d_in[1];
    const float* Wh   = (const float*)d_in[2];
    const float* Wt   = (const float*)d_in[3];
    const float* Wr   = (const float*)d_in[4];
    const float* Wfc  = (const float*)d_in[5];
    const float* bfc  = (const float*)d_in[6];
    const int*   src  = (const int*)d_in[7];
    const int*   dst  = (const int*)d_in[8];

    const int N = in_sizes[0] / DM;   // 32768
    const int E = in_sizes[7];        // 1048576
    float* out = (float*)d_out;

    // workspace layout (floats)
    float* W = (float*)d_ws;
    float* eh   = W;                         // N
    float* et   = eh + N;                    // N
    unsigned int* maxb = (unsigned int*)(et + N);  // N
    float* den  = (float*)(maxb + N);        // N
    float* eatt = den + N;                   // E (logits, then exp in-place)
    float* ft   = eatt + E;                  // N*128 (reused as P)
    float* S    = ft + (size_t)N * DM;       // N*128

    // zero max+denom (0 bits == 0.0f) and ft accumulator
    hipMemsetAsync(maxb, 0, (size_t)2 * N * sizeof(float), stream);
    hipMemsetAsync(ft, 0, (size_t)N * DM * sizeof(float), stream);

    node_scores_kernel<<<N / 8, 256, 0, stream>>>(h, Wh, Wt, eh, et, N);
    edge_scores_kernel<<<E / 8, 256, 0, stream>>>(r, Wr, eh, et, src, dst, eatt, E);
    seg_max_kernel<<<E / 256, 256, 0, stream>>>(eatt, dst, maxb, E);
    exp_denom_kernel<<<E / 256, 256, 0, stream>>>(eatt, dst, maxb, den, E);
    ft_scatter_kernel<<<E / 8, 256, 0, stream>>>(h, eatt, den, src, dst, ft, E);
    z_softmax_kernel<<<N / 16, 256, 0, stream>>>(ft, Wfc, bfc, S, N);

    // reuse ft buffer as P accumulator
    hipMemsetAsync(ft, 0, (size_t)N * DM * sizeof(float), stream);
    p_scatter_kernel<<<E / 8, 256, 0, stream>>>(S, src, dst, ft, E);

    // zero block-diagonal adj region, then write pooled GEMM outputs directly
    hipMemsetAsync(out, 0, (size_t)NKTOT * NKTOT * sizeof(float), stream);
    pool_gemm_kernel<<<BG * 8 * 2, 256, 0, stream>>>(S, ft, h, out);
}